// StochasticRegionalConvolution_73297911873609
// MI455X (gfx1250) — compile-verified
//
#include <hip/hip_runtime.h>

// CDNA5 / gfx1250: wave32, WMMA f32 16x16x4, async global->LDS staging.
typedef __attribute__((ext_vector_type(2))) float v2f;
typedef __attribute__((ext_vector_type(8))) float v8f;

#define WS_ELEMS (9 * 64 * 66)     // ws[tap][co][ci(pad 66)]
#define XS_ELEMS (6 * 66 * 66)     // xs[row 0..5][w 0..65][ci(pad 66)]
#define SL_ELEMS (256)             // S map: 4 rows x 64 cols

// One-instruction gather/scatter: global -> LDS via the async engine (ASYNCcnt).
// LDS dest is a per-lane VGPR byte address (low 32 bits of a generic __shared__ ptr).
__device__ __forceinline__ void async_g2l_b32(const float* gp, float* lp) {
    unsigned lds = (unsigned)(unsigned long long)lp;
    asm volatile("global_load_async_to_lds_b32 %0, %1, off"
                 :: "v"(lds), "v"(gp) : "memory");
}

__global__ void StochasticRegionalConvolution_wmma_kernel(
    const float* __restrict__ x,      // [16][64][256][256]
    const float* __restrict__ wgt,    // [64][64][3][3]
    const int*   __restrict__ h_idx,  // [32]
    const int*   __restrict__ w_idx,  // [32]
    const float* __restrict__ lam,    // [32]
    float* __restrict__ out)          // [16][64][256][256]
{
    extern __shared__ float smem[];
    float* ws = smem;                    // weights, tap-major, ci padded to 66
    float* xs = smem + WS_ELEMS;         // x tile, ci innermost padded to 66
    float* sl = xs + XS_ELEMS;           // coverage-scale map

    const int tid = threadIdx.x;
    const int b   = blockIdx.z;          // batch
    const int h0  = blockIdx.y * 4;      // 4 output rows per block
    const int w0  = blockIdx.x * 64;     // 64 output cols per block

    // ---- stage weights into LDS (async): ws[(kk*64+co)*66 + ci] = wgt[co][ci][kh][kw]
    // 9*64*64 = 144 * 256 exactly -> no divergent tail.
    for (int i = tid; i < 9 * 64 * 64; i += 256) {
        int kk = i % 9;                  // kh*3+kw
        int t  = i / 9;
        int ci = t & 63;
        int co = t >> 6;
        async_g2l_b32(wgt + i, ws + (kk * 64 + co) * 66 + ci);
    }

    // ---- stage x tile into LDS (async, transposed to ci-innermost), clamped at edges
    // 6*64*66 = 99 * 256 exactly -> no divergent tail.
    for (int e = tid; e < 6 * 64 * 66; e += 256) {
        int ww = e % 66;
        int t  = e / 66;
        int ci = t & 63;
        int r  = t >> 6;
        int hg = h0 + r;  if (hg > 255) hg = 255;
        int wg = w0 + ww; if (wg > 255) wg = 255;
        async_g2l_b32(x + ((((b * 64 + ci) * 256) + hg) * 256 + wg),
                      xs + (r * 66 + ww) * 66 + ci);
    }

    // ---- coverage-scale map S(h,w)/16 for this block's 4x64 pixels (regular loads)
    {
        int lr = tid >> 6, ww = tid & 63;
        int hg = h0 + lr, wg = w0 + ww;
        float s = 0.0f;
        for (int t = 0; t < 32; ++t) {
            int dh = hg - h_idx[t];
            int dw = wg - w_idx[t];
            if (dh >= 0 && dh < 62 && dw >= 0 && dw < 62) s += lam[t];
        }
        sl[tid] = s * 0.0625f;           // / n_r (=16)
    }

    // Drain this wave's async copies, then make LDS visible to all waves.
    asm volatile("s_wait_asynccnt 0x0" ::: "memory");
    __syncthreads();

    const int lane   = tid & 31;
    const int wave   = tid >> 5;         // 8 waves
    const int cobase = (wave & 3) * 16;  // which 16 output channels
    const int lrbase = (wave >> 2) * 2;  // which pair of output rows
    const int n      = lane & 15;        // N column / A row within tile
    const int koff   = (lane >> 4) << 1; // 0 (lanes 0-15) or 2 (lanes 16-31)

    v8f acc[8];
    const v8f vzero = {0.f, 0.f, 0.f, 0.f, 0.f, 0.f, 0.f, 0.f};
#pragma unroll
    for (int t = 0; t < 8; ++t) acc[t] = vzero;

    // ---- 9 GEMMs (one per 3x3 tap), K = Cin = 64 in steps of 4
    for (int kh = 0; kh < 3; ++kh) {
        for (int kw = 0; kw < 3; ++kw) {
            const int kk = kh * 3 + kw;
            // A fragment base: row co = cobase + n, K = koff + {0,1}
            const float* ap = ws + (kk * 64 + cobase + n) * 66 + koff;
            // B fragment base: row (lrbase+kh), col (n+kw), K(ci) = koff + {0,1}
            const float* bp = xs + ((lrbase + kh) * 66 + (n + kw)) * 66 + koff;
#pragma unroll
            for (int kc = 0; kc < 16; ++kc) {
                v2f a = *(const v2f*)(ap + 4 * kc);
#pragma unroll
                for (int t = 0; t < 8; ++t) {
                    // tile t: +row (t>>2) => +66*66 dwords; +16 cols (t&3) => +16*66
                    v2f bb = *(const v2f*)(bp + 4 * kc + (t >> 2) * 4356 + (t & 3) * 1056);
                    acc[t] = __builtin_amdgcn_wmma_f32_16x16x4_f32(
                        false, a, false, bb, (short)0, acc[t], false, false);
                }
            }
        }
    }

    // ---- scale by coverage map and store (covers every output pixel)
    const int cohalf = (lane >> 4) * 8;  // C/D layout: lanes 16-31 hold M = v+8
#pragma unroll
    for (int t = 0; t < 8; ++t) {
        int lr = lrbase + (t >> 2);
        int wl = (t & 3) * 16 + n;
        int hg = h0 + lr;
        int wg = w0 + wl;
        float s = sl[lr * 64 + wl];
#pragma unroll
        for (int v = 0; v < 8; ++v) {
            int co = cobase + cohalf + v;
            out[(((long)(b * 64 + co)) * 256 + hg) * 256 + wg] = acc[t][v] * s;
        }
    }
}

extern "C" void kernel_launch(void* const* d_in, const int* in_sizes, int n_in,
                              void* d_out, int out_size, void* d_ws, size_t ws_size,
                              hipStream_t stream) {
    const float* x   = (const float*)d_in[0];
    const float* wgt = (const float*)d_in[1];
    const int*   hi  = (const int*)d_in[2];
    const int*   wi  = (const int*)d_in[3];
    const float* lam = (const float*)d_in[4];
    float* out = (float*)d_out;

    dim3 grid(4, 64, 16);                // w-tiles, h-groups(4 rows), batch
    size_t shmem = (size_t)(WS_ELEMS + XS_ELEMS + SL_ELEMS) * sizeof(float);
    StochasticRegionalConvolution_wmma_kernel<<<grid, 256, shmem, stream>>>(
        x, wgt, hi, wi, lam, out);
}